// VariationalQuantumCircuit_17300128269061
// MI455X (gfx1250) — compile-verified
//
#include <hip/hip_runtime.h>
#include <math.h>

// ---------------------------------------------------------------------------
// Variational quantum circuit, analytically collapsed:
//   * RX/RY layer on |0> => product state: probs factorize per qubit
//   * CNOT chain        => bit relabeling z_0=y_0, z_q = y_q ^ y_{q-1}
//   * RZ layer          => pure phases, cancel under |.|^2
// => probs[b][idx] = prod_q P_q(z_q(idx)) = TA[idx>>4] * TB[idx&31]
// Each 16x16 output tile is an exact rank-1 fp32 outer product, computed with
// V_WMMA_F32_16X16X4_F32 (CDNA5 matrix pipe). Kernel is store-bound:
// 64 MiB out / 23.3 TB/s ~ 2.9 us floor; everything else is preamble.
// ---------------------------------------------------------------------------

typedef __attribute__((ext_vector_type(2))) float v2f;
typedef __attribute__((ext_vector_type(8))) float v8f;

#define NQ 10
#define DIM 1024          // 2^NQ
#define WPB 8             // waves per 256-thread block (wave32)

__global__ __launch_bounds__(256) void vqc_probs_kernel(
    const float* __restrict__ params,  // [nb, 3*NQ]
    float* __restrict__ out,           // [nb, DIM]
    int nb) {
  const int lane = threadIdx.x & 31;
  const int wave = threadIdx.x >> 5;
  const int b = blockIdx.x * WPB + wave;

  // Per-wave LDS region: PQ[20] | TA[64] | TB[32] (padded to 128 floats).
  // TA starts at byte offset 80 within the region -> 8B aligned for b64 loads.
  __shared__ float lds[WPB][128];
  float* PQ = &lds[wave][0];
  float* TA = &lds[wave][20];
  float* TB = &lds[wave][84];

  // ---- Step 1: lanes 0..9 compute per-qubit probabilities P_q(0/1) ----
  if (b < nb && lane < NQ) {
    const int q = lane;
    const float p0 = params[b * (3 * NQ) + 2 * q];      // RX angle
    const float p1 = params[b * (3 * NQ) + 2 * q + 1];  // RY angle
    float cx, sx, cy, sy;
    __sincosf(0.5f * p0, &sx, &cx);
    __sincosf(0.5f * p1, &sy, &cy);
    const float cx2 = cx * cx, sx2 = sx * sx;
    const float cy2 = cy * cy, sy2 = sy * sy;
    v2f pr;
    pr.x = cy2 * cx2 + sy2 * sx2;  // P_q(bit=0)
    pr.y = sy2 * cx2 + cy2 * sx2;  // P_q(bit=1)
    *(v2f*)&PQ[2 * q] = pr;        // one ds_store_b64
  }
  __syncthreads();

  // ---- Step 2: build factor tables ----
  // y_q = bit (9-q) of idx; z_0 = y_0, z_q = y_q ^ y_{q-1}
  // TA[t], t = idx>>4 (qubits 0..5); TB[u], u = idx&31 (qubits 6..9).
  // Entries t=lane and t=lane+32 differ only in bit 5 => z_0 and z_1 flip,
  // factors q=2..5 are shared.
  if (b < nb) {
    const int t = lane;
    float common = 1.0f;
#pragma unroll
    for (int q = 2; q <= 5; ++q) {
      const int z = ((t >> (5 - q)) ^ (t >> (6 - q))) & 1;
      common *= PQ[2 * q + z];
    }
    const int z0 = (t >> 5) & 1;               // 0 for t < 32
    const int z1 = ((t >> 4) ^ (t >> 5)) & 1;  // bit 4 for t < 32
    TA[t]      = common * PQ[z0]     * PQ[2 + z1];
    TA[t + 32] = common * PQ[z0 ^ 1] * PQ[2 + (z1 ^ 1)];

    const int u = lane;
    float v = 1.0f;
#pragma unroll
    for (int q = 6; q <= 9; ++q) {
      const int z = ((u >> (9 - q)) ^ (u >> (10 - q))) & 1;
      v *= PQ[2 * q + z];
    }
    TB[u] = v;
  }
  __syncthreads();

  if (b >= nb) return;

  // ---- Step 3: 4 rank-1 outer products via V_WMMA_F32_16X16X4_F32 ----
  // out[hi][lo] = TA[hi*2 + (lo>>4)] * TB[lo], hi,lo in [0,32).
  // A-matrix (16x4 f32): lanes 0..15 hold row M=lane, v0=K0 (data), v1=K1=0;
  //                      lanes 16..31 hold K2,K3 (zeroed).
  // B-matrix (4x16 f32): lanes 0..15 hold col N=lane, v0=row K0 (data).
  // LDS reads are unconditional (li-address valid for all lanes); lanes
  // 16..31 select 0 via cndmask — no EXEC save/restore around ds_loads.
  const int half = lane >> 4;  // 0: lanes 0-15, 1: lanes 16-31
  const int li = lane & 15;

  const float tb0 = TB[li];        // nh = 0 column factor
  const float tb1 = TB[16 + li];   // nh = 1 column factor
  const float bsel[2] = {half ? 0.0f : tb0, half ? 0.0f : tb1};

  float* const obase = out + (size_t)b * DIM;

#pragma unroll
  for (int mh = 0; mh < 2; ++mh) {
    // TA[(16*mh+li)*2 + {0,1}] are adjacent -> single ds_load_b64
    const v2f tap = *(const v2f*)&TA[(16 * mh + li) * 2];
    const float asel[2] = {half ? 0.0f : tap.x, half ? 0.0f : tap.y};
#pragma unroll
    for (int nh = 0; nh < 2; ++nh) {
      v2f av;
      av.x = asel[nh];
      av.y = 0.0f;
      v2f bv;
      bv.x = bsel[nh];
      bv.y = 0.0f;
      v8f c = {};
      // 8 args: (neg_a, A, neg_b, B, c_mod, C, reuse_a, reuse_b)
      v8f d = __builtin_amdgcn_wmma_f32_16x16x4_f32(
          false, av, false, bv, (short)0, c, false, false);

      // D layout: VGPR r, lanes 0-15 -> M=r, lanes 16-31 -> M=r+8, N=li.
      // Each clause'd store writes two contiguous 64B segments.
      float* const o = obase + (16 * mh + half * 8) * 32 + nh * 16 + li;
#pragma unroll
      for (int r = 0; r < 8; ++r) {
        o[r * 32] = d[r];
      }
    }
  }
}

extern "C" void kernel_launch(void* const* d_in, const int* in_sizes, int n_in,
                              void* d_out, int out_size, void* d_ws,
                              size_t ws_size, hipStream_t stream) {
  const float* params = (const float*)d_in[0];
  float* out = (float*)d_out;
  const int nb = in_sizes[0] / (3 * NQ);  // batch count (16384)
  const int blocks = (nb + WPB - 1) / WPB;
  vqc_probs_kernel<<<blocks, 256, 0, stream>>>(params, out, nb);
}